// SkipGINNet_83485574300300
// MI455X (gfx1250) — compile-verified
//
#include <hip/hip_runtime.h>
#include <hip/hip_bf16.h>
#include <math.h>

#define NN 100000
#define NE 1600000
#define INF_ 16
#define HIDN 128
#define NL 8

typedef __attribute__((ext_vector_type(2))) float v2f;
typedef __attribute__((ext_vector_type(8))) float v8f;

__device__ __forceinline__ void atomicAddF32(float* p, float v) {
    __hip_atomic_fetch_add(p, v, __ATOMIC_RELAXED, __HIP_MEMORY_SCOPE_AGENT);
}

__global__ void zero_kernel(float* __restrict__ p, long n) {
    long i = (long)blockIdx.x * blockDim.x + threadIdx.x;
    long stride = (long)gridDim.x * blockDim.x;
    for (; i < n; i += stride) p[i] = 0.f;
}

// One wave per edge (F=128) or 8 edges per wave (F=16). Lane handles a float4 chunk.
__global__ void scatter_kernel(const float* __restrict__ h,
                               const int* __restrict__ src,
                               const int* __restrict__ dst,
                               float* __restrict__ agg,
                               int E, int F, int shift) {
    int tid = blockIdx.x * blockDim.x + threadIdx.x;
    int e = tid >> shift;
    if (e >= E) return;
    int chunk = tid & ((1 << shift) - 1);
    int s = src[e], d = dst[e];
    const float4 v = *(const float4*)(h + (long)s * F + chunk * 4);
    float* ap = agg + (long)d * F + chunk * 4;
    atomicAddF32(ap + 0, v.x);
    atomicAddF32(ap + 1, v.y);
    atomicAddF32(ap + 2, v.z);
    atomicAddF32(ap + 3, v.w);
}

// Y[M,128] = T(A)[M,K] @ W[K,128] + bias; also accumulates column sum / sum-of-squares.
// mode 0: T(A) = aw*A0 + A1   (GIN aggregate; aw = 1+eps[layer] read on-device)
// mode 1: T(A) = relu(scale*A0 + shift)   (fused inner BN+ReLU)
// mode 2: T(A) = wconst*A0    (skip average)
__global__ __launch_bounds__(256) void gemm_bn_kernel(
    const float* __restrict__ A0, const float* __restrict__ A1,
    const float* __restrict__ W, const float* __restrict__ bias,
    const float* __restrict__ scale, const float* __restrict__ shift,
    const float* __restrict__ epsPtr, int epsIdx, float wconst,
    int mode, int K, int kshift,
    float* __restrict__ Y, float* __restrict__ colsum,
    float* __restrict__ colsumsq)
{
    __shared__ float As[16 * 130];
    const int lds = K + 2;                 // padded stride: conflict-free frag reads
    const int rowBase = blockIdx.x * 16;

    float aw = wconst;
    if (mode == 0) aw = 1.0f + epsPtr[epsIdx];

    // Stage transformed A slab (16 x K) into LDS
    for (int i = threadIdx.x; i < 16 * K; i += 256) {
        int r = i >> kshift, c = i & (K - 1);
        long g = (long)(rowBase + r) * K + c;
        float v;
        if (mode == 0)      v = aw * A0[g] + A1[g];
        else if (mode == 1) v = fmaxf(scale[c] * A0[g] + shift[c], 0.f);
        else                v = aw * A0[g];
        As[r * lds + c] = v;
    }
    __syncthreads();

    const int lane  = threadIdx.x & 31;
    const int wave  = threadIdx.x >> 5;   // column tile 0..7
    const int laneN = lane & 15;          // N for B/C/D, M for A
    const int half  = lane >> 4;
    const int col   = wave * 16 + laneN;

    v8f acc = {0.f, 0.f, 0.f, 0.f, 0.f, 0.f, 0.f, 0.f};
    for (int k0 = 0; k0 < K; k0 += 4) {
        const int ka = k0 + 2 * half;     // K = vgpr + 2*laneHalf (ISA layout)
        v2f a, b;
        a.x = As[laneN * lds + ka];
        a.y = As[laneN * lds + ka + 1];
        b.x = W[(long)ka * HIDN + col];
        b.y = W[(long)(ka + 1) * HIDN + col];
        acc = __builtin_amdgcn_wmma_f32_16x16x4_f32(
            false, a, false, b, (short)0, acc, false, false);
    }

    const float bcol = bias[col];
    float s = 0.f, q = 0.f;
#pragma unroll
    for (int v = 0; v < 8; v++) {
        int m = rowBase + v + 8 * half;   // C/D layout: M = vgpr + 8*laneHalf
        float y = acc[v] + bcol;
        Y[(long)m * HIDN + col] = y;
        s += y;
        q += y * y;
    }
    s += __shfl_xor(s, 16, 32);
    q += __shfl_xor(q, 16, 32);
    if (half == 0) {
        atomicAddF32(&colsum[col], s);
        atomicAddF32(&colsumsq[col], q);
    }
}

__global__ void stats_kernel(const float* __restrict__ colsum,
                             const float* __restrict__ colsumsq,
                             const float* __restrict__ gamma,
                             const float* __restrict__ beta,
                             float invN, float* __restrict__ scale,
                             float* __restrict__ shift) {
    int i = threadIdx.x;
    if (i < HIDN) {
        float mu  = colsum[i] * invN;
        float var = colsumsq[i] * invN - mu * mu;
        float sc  = gamma[i] * rsqrtf(var + 1e-5f);
        scale[i] = sc;
        shift[i] = beta[i] - mu * sc;
    }
}

__global__ void bnrelu_kernel(const float* __restrict__ y,
                              const float* __restrict__ scale,
                              const float* __restrict__ shift,
                              float* __restrict__ h, float* __restrict__ skip,
                              int doSkip) {
    long idx = (long)blockIdx.x * blockDim.x + threadIdx.x;
    if (idx >= (long)NN * HIDN) return;
    int c = idx & (HIDN - 1);
    float v = fmaxf(scale[c] * y[idx] + shift[c], 0.f);
    h[idx] = v;
    if (doSkip) skip[idx] += v;
}

// One wave per node: relu(bn(yr)) dot Wr2 -> sigmoid
__global__ void head_kernel(const float* __restrict__ yr,
                            const float* __restrict__ scale,
                            const float* __restrict__ shift,
                            const float* __restrict__ Wr2,
                            const float* __restrict__ br2,
                            float* __restrict__ out) {
    int wave = threadIdx.x >> 5;
    int lane = threadIdx.x & 31;
    int node = blockIdx.x * 8 + wave;
    if (node >= NN) return;
    const float* row = yr + (long)node * HIDN;
    float s = 0.f;
#pragma unroll
    for (int j = 0; j < 4; j++) {
        int c = lane + 32 * j;
        float r = fmaxf(scale[c] * row[c] + shift[c], 0.f);
        s += r * Wr2[c];
    }
    for (int off = 16; off > 0; off >>= 1) s += __shfl_xor(s, off, 32);
    if (lane == 0) out[node] = 1.f / (1.f + expf(-(s + br2[0])));
}

extern "C" void kernel_launch(void* const* d_in, const int* in_sizes, int n_in,
                              void* d_out, int out_size, void* d_ws, size_t ws_size,
                              hipStream_t stream) {
    const float* x      = (const float*)d_in[0];
    const int*   ei     = (const int*)d_in[1];
    const float* eps    = (const float*)d_in[2];
    const float* W1_0   = (const float*)d_in[3];
    const float* b1_0   = (const float*)d_in[4];
    const float* W1     = (const float*)d_in[5];
    const float* b1     = (const float*)d_in[6];
    const float* g_in   = (const float*)d_in[7];
    const float* be_in  = (const float*)d_in[8];
    const float* W2     = (const float*)d_in[9];
    const float* b2     = (const float*)d_in[10];
    const float* g_out  = (const float*)d_in[11];
    const float* be_out = (const float*)d_in[12];
    const float* Wr1    = (const float*)d_in[13];
    const float* br1    = (const float*)d_in[14];
    const float* gr     = (const float*)d_in[15];
    const float* ber    = (const float*)d_in[16];
    const float* Wr2    = (const float*)d_in[17];
    const float* br2    = (const float*)d_in[18];
    float* out = (float*)d_out;

    const int* src = ei;
    const int* dst = ei + NE;

    float* ws = (float*)d_ws;
    const long MH = (long)NN * HIDN;
    float* hbuf     = ws;            // h after each layer
    float* aggbuf   = ws + MH;       // agg, then reused as y2
    float* y1buf    = ws + 2 * MH;   // inner-MLP pre-BN output / regressor GEMM out
    float* skip     = ws + 3 * MH;   // skip accumulator
    float* colsum   = ws + 4 * MH;
    float* colsumsq = colsum + HIDN;
    float* scbuf    = colsum + 2 * HIDN;
    float* shbuf    = colsum + 3 * HIDN;

    const dim3 b256(256);
    const float invN = 1.0f / NN;
    const int gemmBlocks = NN / 16;              // 6250
    const int bnBlocks   = (int)((MH + 255) / 256);

    zero_kernel<<<2048, b256, 0, stream>>>(skip, MH);

    for (int i = 0; i < NL; i++) {
        const int F = (i == 0) ? INF_ : HIDN;
        const int kshift = (i == 0) ? 4 : 7;
        const float* hcur = (i == 0) ? x : hbuf;

        zero_kernel<<<2048, b256, 0, stream>>>(aggbuf, (long)NN * F);
        const int esh = (F == 16) ? 2 : 5;
        const long nthr = (long)NE << esh;
        scatter_kernel<<<(int)((nthr + 255) / 256), b256, 0, stream>>>(
            hcur, src, dst, aggbuf, NE, F, esh);

        // GEMM1: y1 = ((1+eps)h + agg) @ Wa + ba   (+ column stats)
        zero_kernel<<<1, b256, 0, stream>>>(colsum, 2 * HIDN);
        const float* Wa = (i == 0) ? W1_0 : (W1 + (long)(i - 1) * HIDN * HIDN);
        const float* ba = (i == 0) ? b1_0 : (b1 + (long)(i - 1) * HIDN);
        gemm_bn_kernel<<<gemmBlocks, b256, 0, stream>>>(
            hcur, aggbuf, Wa, ba, nullptr, nullptr, eps, i, 0.f, 0, F, kshift,
            y1buf, colsum, colsumsq);
        stats_kernel<<<1, 128, 0, stream>>>(colsum, colsumsq,
            g_in + i * HIDN, be_in + i * HIDN, invN, scbuf, shbuf);

        // GEMM2: y2 = relu(bn1(y1)) @ W2 + b2   (+ column stats); y2 reuses aggbuf
        zero_kernel<<<1, b256, 0, stream>>>(colsum, 2 * HIDN);
        gemm_bn_kernel<<<gemmBlocks, b256, 0, stream>>>(
            y1buf, nullptr, W2 + (long)i * HIDN * HIDN, b2 + i * HIDN,
            scbuf, shbuf, nullptr, 0, 0.f, 1, HIDN, 7,
            aggbuf, colsum, colsumsq);
        stats_kernel<<<1, 128, 0, stream>>>(colsum, colsumsq,
            g_out + i * HIDN, be_out + i * HIDN, invN, scbuf, shbuf);

        // h = relu(bn2(y2)); skip += h on odd layers
        bnrelu_kernel<<<bnBlocks, b256, 0, stream>>>(
            aggbuf, scbuf, shbuf, hbuf, skip, (i & 1));
    }

    // Regressor: yr = (skip/4) @ Wr1 + br1 (+ stats); out = sigmoid(relu(bn(yr)) @ Wr2 + br2)
    zero_kernel<<<1, b256, 0, stream>>>(colsum, 2 * HIDN);
    gemm_bn_kernel<<<gemmBlocks, b256, 0, stream>>>(
        skip, nullptr, Wr1, br1, nullptr, nullptr, nullptr, 0, 0.25f, 2, HIDN, 7,
        y1buf, colsum, colsumsq);
    stats_kernel<<<1, 128, 0, stream>>>(colsum, colsumsq, gr, ber, invN, scbuf, shbuf);
    head_kernel<<<(NN + 7) / 8, b256, 0, stream>>>(y1buf, scbuf, shbuf, Wr2, br2, out);
}